// GCN_75453985456218
// MI455X (gfx1250) — compile-verified
//
#include <hip/hip_runtime.h>
#include <hip/hip_bf16.h>
#include <math.h>

typedef __attribute__((ext_vector_type(2))) float v2f;
typedef __attribute__((ext_vector_type(8))) float v8f;

#define N_FEAT 300
#define HID    32
#define NC     10
#define NCP    16   // classes padded to 16 for WMMA tiles

// ---------------------------------------------------------------------------
// Degree / normalization
// ---------------------------------------------------------------------------
__global__ void init_deg_kernel(float* __restrict__ deg, int N) {
    int i = blockIdx.x * blockDim.x + threadIdx.x;
    if (i < N) deg[i] = 1.0f;  // self-loop contributes 1 to every node's degree
}

__global__ void deg_accum_kernel(const int* __restrict__ dst,
                                 float* __restrict__ deg, int E) {
    int e = blockIdx.x * blockDim.x + threadIdx.x;
    if (e < E) unsafeAtomicAdd(&deg[dst[e]], 1.0f);  // global_atomic_add_f32
}

__global__ void dinv_kernel(float* __restrict__ deg, int N) {
    int i = blockIdx.x * blockDim.x + threadIdx.x;
    if (i < N) {
        float d = deg[i];
        deg[i] = (d > 0.0f) ? rsqrtf(d) : 0.0f;   // in place: deg -> deg^-1/2
    }
}

// ---------------------------------------------------------------------------
// GEMM1: H1raw[N,32] = X[N,300] @ W1[300,32]   (fp32 WMMA 16x16x4)
// One wave -> 16 rows x 32 cols (two 16x16 accumulators, shared A fragment).
// Wave index forced into SGPRs via readfirstlane so tile guards are scalar.
// A layout (16x4 f32): lanes 0-15 = rows, v0/v1 hold K=0/1; lanes 16-31 K=2/3.
// B layout (4x16): lanes 0-15 = cols, v0/v1 hold K=0/1; lanes 16-31 K=2/3.
// ---------------------------------------------------------------------------
__global__ void gemm1_wmma_kernel(const float* __restrict__ X,
                                  const float* __restrict__ W1,
                                  float* __restrict__ H1raw, int N) {
    const int lane = threadIdx.x & 31;
    const int wave = __builtin_amdgcn_readfirstlane(
        (int)(blockIdx.x * (blockDim.x >> 5) + (threadIdx.x >> 5)));  // SGPR
    const int mtiles = (N + 15) >> 4;
    if (wave >= mtiles) return;          // scalar branch

    const int m0   = wave << 4;          // SGPR
    const int half = lane >> 4;          // 0: K pair {0,1}; 1: K pair {2,3}
    const int l    = lane & 15;
    int row = m0 + l;
    if (row >= N) row = N - 1;           // clamp (tail stores are guarded)
    const int koff = half * 2;

    const float* xrow = X + (size_t)row * N_FEAT + koff;
    v8f c0 = {}; v8f c1 = {};

    for (int k = 0; k < N_FEAT; k += 4) {
        v2f a;
        a.x = xrow[k];
        a.y = xrow[k + 1];
        const float* w = W1 + (size_t)(k + koff) * HID + l;
        v2f b0, b1;
        b0.x = w[0];   b0.y = w[HID];        // cols 0..15
        b1.x = w[16];  b1.y = w[HID + 16];   // cols 16..31
        c0 = __builtin_amdgcn_wmma_f32_16x16x4_f32(false, a, false, b0,
                                                   (short)0, c0, false, false);
        c1 = __builtin_amdgcn_wmma_f32_16x16x4_f32(false, a, false, b1,
                                                   (short)0, c1, false, false);
    }

    const int mbase = m0 + half * 8;  // C/D: lanes>=16 hold M rows +8
    if (m0 + 15 < N) {
        // scalar full-tile test -> straight-line store clause
        float* o = H1raw + (size_t)mbase * HID + l;
#pragma unroll
        for (int v = 0; v < 8; ++v) {
            o[(size_t)v * HID]      = c0[v];
            o[(size_t)v * HID + 16] = c1[v];
        }
    } else {
#pragma unroll
        for (int v = 0; v < 8; ++v) {
            int m = mbase + v;
            if (m < N) {
                H1raw[(size_t)m * HID + l]      = c0[v];
                H1raw[(size_t)m * HID + 16 + l] = c1[v];
            }
        }
    }
}

// ---------------------------------------------------------------------------
// Layer-1 propagation: one wave per edge, one feature per lane.
// Edge metadata is wave-uniform -> force scalar path via readfirstlane.
// H1agg[dst,j] += H1raw[src,j] * dinv[src]*dinv[dst]
// ---------------------------------------------------------------------------
__global__ void prop1_kernel(const int* __restrict__ src,
                             const int* __restrict__ dst,
                             const float* __restrict__ dinv,
                             const float* __restrict__ H1raw,
                             float* __restrict__ H1agg, int E) {
    unsigned t = blockIdx.x * blockDim.x + threadIdx.x;
    int e = __builtin_amdgcn_readfirstlane((int)(t >> 5));  // uniform per wave
    int j = threadIdx.x & 31;
    if (e >= E) return;
    int s = src[e], d = dst[e];                 // scalar loads
    float nrm = dinv[s] * dinv[d];              // scalar loads
    unsafeAtomicAdd(&H1agg[(size_t)d * HID + j],
                    H1raw[(size_t)s * HID + j] * nrm);
}

// Self-loop term + bias + tanh (in place on H1agg -> activation)
__global__ void act1_kernel(float* __restrict__ H1agg,
                            const float* __restrict__ H1raw,
                            const float* __restrict__ dinv,
                            const float* __restrict__ b1, int N) {
    int t = blockIdx.x * blockDim.x + threadIdx.x;
    if (t >= N * HID) return;
    int i = t >> 5;
    int j = t & 31;
    float di = dinv[i];
    float v = H1agg[t] + H1raw[t] * di * di + b1[j];
    H1agg[t] = tanhf(v);
}

// ---------------------------------------------------------------------------
// GEMM2: H2raw[N,16] = H1act[N,32] @ W2pad[32,16]  (cols 10..15 are zero)
// ---------------------------------------------------------------------------
__global__ void gemm2_wmma_kernel(const float* __restrict__ H1act,
                                  const float* __restrict__ W2,
                                  float* __restrict__ H2raw, int N) {
    const int lane = threadIdx.x & 31;
    const int wave = __builtin_amdgcn_readfirstlane(
        (int)(blockIdx.x * (blockDim.x >> 5) + (threadIdx.x >> 5)));  // SGPR
    const int mtiles = (N + 15) >> 4;
    if (wave >= mtiles) return;          // scalar branch

    const int m0   = wave << 4;          // SGPR
    const int half = lane >> 4;
    const int l    = lane & 15;
    int row = m0 + l;
    if (row >= N) row = N - 1;
    const int koff = half * 2;

    const float* hrow = H1act + (size_t)row * HID + koff;
    v8f c = {};

#pragma unroll
    for (int k = 0; k < HID; k += 4) {
        v2f a;
        a.x = hrow[k];
        a.y = hrow[k + 1];
        int kk = k + koff;
        v2f b;
        b.x = (l < NC) ? W2[(size_t)kk * NC + l] : 0.0f;
        b.y = (l < NC) ? W2[(size_t)(kk + 1) * NC + l] : 0.0f;
        c = __builtin_amdgcn_wmma_f32_16x16x4_f32(false, a, false, b,
                                                  (short)0, c, false, false);
    }

    const int mbase = m0 + half * 8;
    if (m0 + 15 < N) {
        float* o = H2raw + (size_t)mbase * NCP + l;
#pragma unroll
        for (int v = 0; v < 8; ++v)
            o[(size_t)v * NCP] = c[v];
    } else {
#pragma unroll
        for (int v = 0; v < 8; ++v) {
            int m = mbase + v;
            if (m < N) H2raw[(size_t)m * NCP + l] = c[v];
        }
    }
}

// ---------------------------------------------------------------------------
// Layer-2 propagation: 16 lanes per edge, 10 active channels.
// ---------------------------------------------------------------------------
__global__ void prop2_kernel(const int* __restrict__ src,
                             const int* __restrict__ dst,
                             const float* __restrict__ dinv,
                             const float* __restrict__ H2raw,
                             float* __restrict__ L2agg, int E) {
    unsigned t = blockIdx.x * blockDim.x + threadIdx.x;
    int e = t >> 4;
    int j = t & 15;
    if (e >= E || j >= NC) return;
    int s = src[e], d = dst[e];
    float nrm = dinv[s] * dinv[d];
    unsafeAtomicAdd(&L2agg[(size_t)d * NCP + j],
                    H2raw[(size_t)s * NCP + j] * nrm);
}

// Self-loop + bias + log_softmax -> d_out[N,10]
__global__ void finalize_kernel(const float* __restrict__ L2agg,
                                const float* __restrict__ H2raw,
                                const float* __restrict__ dinv,
                                const float* __restrict__ b2,
                                float* __restrict__ out, int N) {
    int i = blockIdx.x * blockDim.x + threadIdx.x;
    if (i >= N) return;
    float di = dinv[i];
    float d2 = di * di;
    float l[NC];
    float mx = -3.402823466e38f;
#pragma unroll
    for (int j = 0; j < NC; ++j) {
        float v = L2agg[(size_t)i * NCP + j] + H2raw[(size_t)i * NCP + j] * d2 + b2[j];
        l[j] = v;
        mx = fmaxf(mx, v);
    }
    float s = 0.0f;
#pragma unroll
    for (int j = 0; j < NC; ++j) s += expf(l[j] - mx);
    float ls = logf(s);
#pragma unroll
    for (int j = 0; j < NC; ++j) out[(size_t)i * NC + j] = l[j] - mx - ls;
}

// ---------------------------------------------------------------------------
extern "C" void kernel_launch(void* const* d_in, const int* in_sizes, int n_in,
                              void* d_out, int out_size, void* d_ws, size_t ws_size,
                              hipStream_t stream) {
    const float* X  = (const float*)d_in[0];
    const int*   ei = (const int*)d_in[1];
    const float* W1 = (const float*)d_in[2];
    const float* b1 = (const float*)d_in[3];
    const float* W2 = (const float*)d_in[4];
    const float* b2 = (const float*)d_in[5];
    float* out = (float*)d_out;

    const int N = in_sizes[0] / N_FEAT;
    const int E = in_sizes[1] / 2;
    const int* src = ei;        // edge_index[0]
    const int* dst = ei + E;    // edge_index[1]

    float* ws    = (float*)d_ws;
    float* deg   = ws;                               // N       (becomes dinv)
    float* H1raw = deg   + (size_t)N;                // N*32
    float* H1agg = H1raw + (size_t)N * HID;          // N*32    (becomes H1act)
    float* H2raw = H1agg + (size_t)N * HID;          // N*16
    float* L2agg = H2raw + (size_t)N * NCP;          // N*16

    hipMemsetAsync(H1agg, 0, (size_t)N * HID * sizeof(float), stream);
    hipMemsetAsync(L2agg, 0, (size_t)N * NCP * sizeof(float), stream);

    const int T = 256;
    init_deg_kernel<<<(N + T - 1) / T, T, 0, stream>>>(deg, N);
    deg_accum_kernel<<<(E + T - 1) / T, T, 0, stream>>>(dst, deg, E);
    dinv_kernel<<<(N + T - 1) / T, T, 0, stream>>>(deg, N);

    const int mtiles = (N + 15) / 16;
    const int gemmBlocks = (mtiles + 7) / 8;   // 8 waves per 256-thread block
    gemm1_wmma_kernel<<<gemmBlocks, T, 0, stream>>>(X, W1, H1raw, N);

    long long p1t = (long long)E * 32;
    prop1_kernel<<<(int)((p1t + T - 1) / T), T, 0, stream>>>(src, dst, deg, H1raw, H1agg, E);

    act1_kernel<<<(N * HID + T - 1) / T, T, 0, stream>>>(H1agg, H1raw, deg, b1, N);

    gemm2_wmma_kernel<<<gemmBlocks, T, 0, stream>>>(H1agg, W2, H2raw, N);

    long long p2t = (long long)E * 16;
    prop2_kernel<<<(int)((p2t + T - 1) / T), T, 0, stream>>>(src, dst, deg, H2raw, L2agg, E);

    finalize_kernel<<<(N + T - 1) / T, T, 0, stream>>>(L2agg, H2raw, deg, b2, out, N);
}